// Trittention_pytorch_46823733461713
// MI455X (gfx1250) — compile-verified
//
#include <hip/hip_runtime.h>
#include <hip/hip_bf16.h>

// ---------------------------------------------------------------------------
// Trittention forward for MI455X (gfx1250, wave32, WMMA).
//
// Shapes: b=1, n=8, S=192, h=64, all fp32.
// Outputs (concatenated flat): z[8,192,64], score[8,192,192,192],
//                              v_gated[8,192,192,64], M[8,192].
// ~302MB of output => store-bandwidth bound (~13us floor @ 23.3TB/s).
// Matrix contractions use V_WMMA_F32_16X16X4_F32 (fp32 in/out, no downconvert).
// Softmax-weighted value sum is algebraically factorized over s then t,
// turning the O(S^3*H) contraction into two 192x192x64 WMMA GEMMs per head.
// The two write-once streaming outputs use non-temporal (TH=NT) b128 stores so
// the 302MB stream does not evict the ~5MB reused workspace from the 192MB L2.
// ---------------------------------------------------------------------------

typedef __attribute__((ext_vector_type(2))) float v2f;
typedef __attribute__((ext_vector_type(4))) float v4f;
typedef __attribute__((ext_vector_type(8))) float v8f;

#define SQ 192          // sequence length
#define HD 64           // head dim
#define NH 8            // heads
#define MASK_FILL (-1000000.0f)
#define SCALE 0.125f

static __device__ __forceinline__ v8f wmma4(v2f a, v2f b, v8f c) {
  // V_WMMA_F32_16X16X4_F32: A 16x4 f32 (2 VGPR/lane), B 4x16 f32, C/D 16x16 f32.
  return __builtin_amdgcn_wmma_f32_16x16x4_f32(false, a, false, b, (short)0, c,
                                               false, false);
}

// ---------------------------------------------------------------------------
// Kernel 1: per-head 192x192 = [192x64] . [192x64]^T GEMMs via WMMA.
//   which=0: qk[t,q] = k2[t] . q[q]     which=1: kk[s,t] = k1[s] . k2[t]
// One wave per 16x16 C tile; K=64 -> 16 WMMA f32 16x16x4 ops.
// A lane layout: lane L holds A[M=L&15][K = k0 + 2*(L>>4) .. +1] (ISA 7.12.2).
// B lane layout: lane L holds B[K = k0 + 2*(L>>4) .. +1][N=L&15].
// ---------------------------------------------------------------------------
__global__ __launch_bounds__(32) void k_gemm_scores(
    const float* __restrict__ q, const float* __restrict__ k1,
    const float* __restrict__ k2, float* __restrict__ qk,
    float* __restrict__ kk) {
  const int tm = blockIdx.x, tn = blockIdx.y;
  const int head = blockIdx.z >> 1, which = blockIdx.z & 1;
  const size_t hin = (size_t)head * SQ * HD;
  const float* A = which ? (k1 + hin) : (k2 + hin);
  const float* B = which ? (k2 + hin) : (q + hin);
  float* C = (which ? kk : qk) + (size_t)head * SQ * SQ;

  const int lane = threadIdx.x, half = lane >> 4, l15 = lane & 15;
  const int kb = half * 2;
  const float* arow = A + (size_t)(tm * 16 + l15) * HD + kb;
  const float* brow = B + (size_t)(tn * 16 + l15) * HD + kb;

  v8f c = {};
#pragma unroll
  for (int k0 = 0; k0 < HD; k0 += 4) {
    v2f a = *(const v2f*)(arow + k0);  // global_load_b64
    v2f b = *(const v2f*)(brow + k0);
    c = wmma4(a, b, c);
  }
  // C tile: VGPR r -> row M = r + 8*half, col N = l15.
  float* crow = C + (size_t)(tm * 16 + half * 8) * SQ + tn * 16 + l15;
#pragma unroll
  for (int r = 0; r < 8; ++r) crow[(size_t)r * SQ] = c[r];
}

// ---------------------------------------------------------------------------
// Kernel 2: per-head softmax statistics + silu(v1) + M output.
//   mkk[t] = max_{s<=t} 0.125*kk[s,t]        Ekk[t] = sum_{s<=t} exp(. - mkk)
//   M0[q]  = max_{t<=q} (0.125*qk[t,q]+mkk)  SE[q]  = sum_{t<=q} exp(.-M0)*Ekk
//   M[q]   = M0[q] + log(SE[q] + 0.01)
// ---------------------------------------------------------------------------
__global__ __launch_bounds__(192) void k_stats(
    const float* __restrict__ v1, const float* __restrict__ qk,
    const float* __restrict__ kk, float* __restrict__ SV,
    float* __restrict__ mkk_g, float* __restrict__ M0_g,
    float* __restrict__ SE_g, float* __restrict__ M_out) {
  const int head = blockIdx.x, tid = threadIdx.x;
  const float* kkh = kk + (size_t)head * SQ * SQ;
  const float* qkh = qk + (size_t)head * SQ * SQ;
  __shared__ float s_m[SQ];
  __shared__ float s_e[SQ];

  // silu(v1) staged to workspace for the W-GEMM and v_gated writer.
  const float* v1h = v1 + (size_t)head * SQ * HD;
  float* SVh = SV + (size_t)head * SQ * HD;
  for (int i = tid; i < SQ * HD; i += 192) {
    float x = v1h[i];
    SVh[i] = x / (1.0f + __expf(-x));
  }

  {  // column stats of kk (thread t owns column t)
    const int t = tid;
    float m = -3.4e38f;
    for (int s = 0; s <= t; ++s) m = fmaxf(m, SCALE * kkh[s * SQ + t]);
    float e = 0.0f;
    for (int s = 0; s <= t; ++s) e += __expf(SCALE * kkh[s * SQ + t] - m);
    s_m[t] = m;
    s_e[t] = e;
    mkk_g[head * SQ + t] = m;
  }
  __syncthreads();
  {  // per-query stats (thread qq owns query qq)
    const int qq = tid;
    float m0 = -3.4e38f;
    for (int t = 0; t <= qq; ++t)
      m0 = fmaxf(m0, SCALE * qkh[t * SQ + qq] + s_m[t]);
    float se = 0.0f;
    for (int t = 0; t <= qq; ++t)
      se += __expf(SCALE * qkh[t * SQ + qq] + s_m[t] - m0) * s_e[t];
    M0_g[head * SQ + qq] = m0;
    SE_g[head * SQ + qq] = se;
    M_out[head * SQ + qq] = m0 + __logf(se + 0.01f);
  }
}

// ---------------------------------------------------------------------------
// Kernel 3: materialize the two exp-factor matrices per head.
//   E[s,t] = (s<=t) ? exp(0.125*kk[s,t] - mkk[t])            : 0
//   P[q,t] = (t<=q) ? exp(0.125*qk[t,q] + mkk[t] - M0[q])    : 0
// ---------------------------------------------------------------------------
__global__ __launch_bounds__(256) void k_pe(
    const float* __restrict__ qk, const float* __restrict__ kk,
    const float* __restrict__ mkk_g, const float* __restrict__ M0_g,
    float* __restrict__ E, float* __restrict__ P) {
  const int head = blockIdx.x, tid = threadIdx.x;
  __shared__ float s_m[SQ];
  __shared__ float s_m0[SQ];
  if (tid < SQ) {
    s_m[tid] = mkk_g[head * SQ + tid];
    s_m0[tid] = M0_g[head * SQ + tid];
  }
  __syncthreads();
  const float* kkh = kk + (size_t)head * SQ * SQ;
  const float* qkh = qk + (size_t)head * SQ * SQ;
  float* Eh = E + (size_t)head * SQ * SQ;
  float* Ph = P + (size_t)head * SQ * SQ;
  for (int idx = tid; idx < SQ * SQ; idx += 256) {
    const int r = idx / SQ, t = idx - r * SQ;
    Eh[idx] = (r <= t) ? __expf(SCALE * kkh[idx] - s_m[t]) : 0.0f;
    Ph[idx] = (t <= r) ? __expf(SCALE * qkh[t * SQ + r] + s_m[t] - s_m0[r]) : 0.0f;
  }
}

// ---------------------------------------------------------------------------
// Kernel 4: G[t,h] = v2[t,h] * (E^T . silu_v1)[t,h]   (WMMA, K=192 -> 48 ops)
//   A[t][s] = E[s,t] (transposed reads), B[s][h] = silu_v1.
// ---------------------------------------------------------------------------
__global__ __launch_bounds__(32) void k_gemm_w(
    const float* __restrict__ E, const float* __restrict__ SV,
    const float* __restrict__ v2, float* __restrict__ G) {
  const int tt = blockIdx.x, th = blockIdx.y, head = blockIdx.z;
  const float* Eh = E + (size_t)head * SQ * SQ;
  const float* SVh = SV + (size_t)head * SQ * HD;
  const float* v2h = v2 + (size_t)head * SQ * HD;
  float* Gh = G + (size_t)head * SQ * HD;

  const int lane = threadIdx.x, half = lane >> 4, l15 = lane & 15;
  const int kb = half * 2;
  const int trow = tt * 16 + l15;
  const int hcol = th * 16 + l15;

  v8f c = {};
  for (int k0 = 0; k0 < SQ; k0 += 4) {
    v2f a, b;
    a.x = Eh[(k0 + kb) * SQ + trow];       // A[t][s] = E[s][t]
    a.y = Eh[(k0 + kb + 1) * SQ + trow];
    b.x = SVh[(k0 + kb) * HD + hcol];      // B[s][h]
    b.y = SVh[(k0 + kb + 1) * HD + hcol];
    c = wmma4(a, b, c);
  }
#pragma unroll
  for (int r = 0; r < 8; ++r) {
    const int t = tt * 16 + half * 8 + r;
    Gh[t * HD + hcol] = c[r] * v2h[t * HD + hcol];
  }
}

// ---------------------------------------------------------------------------
// Kernel 5: z[q,h] = (P . G)[q,h] / SE[q]   (WMMA, K=192 -> 48 ops)
// ---------------------------------------------------------------------------
__global__ __launch_bounds__(32) void k_gemm_z(
    const float* __restrict__ P, const float* __restrict__ G,
    const float* __restrict__ SE_g, float* __restrict__ z) {
  const int tq = blockIdx.x, th = blockIdx.y, head = blockIdx.z;
  const float* Ph = P + (size_t)head * SQ * SQ;
  const float* Gh = G + (size_t)head * SQ * HD;

  const int lane = threadIdx.x, half = lane >> 4, l15 = lane & 15;
  const int kb = half * 2;
  const float* prow = Ph + (size_t)(tq * 16 + l15) * SQ + kb;
  const int hcol = th * 16 + l15;

  v8f c = {};
  for (int k0 = 0; k0 < SQ; k0 += 4) {
    v2f a = *(const v2f*)(prow + k0);      // contiguous b64 loads
    v2f b;
    b.x = Gh[(k0 + kb) * HD + hcol];
    b.y = Gh[(k0 + kb + 1) * HD + hcol];
    c = wmma4(a, b, c);
  }
#pragma unroll
  for (int r = 0; r < 8; ++r) {
    const int qq = tq * 16 + half * 8 + r;
    z[((size_t)head * SQ + qq) * HD + hcol] = c[r] / SE_g[head * SQ + qq];
  }
}

// ---------------------------------------------------------------------------
// Kernel 6: stream pre_softmax_attn_score (226 MB), non-temporal b128 stores.
//   out[n,s,t,q] = (s>t || t>q) ? -1e6 : kk[s,t] + qk[t,q]
// One block per (n,s); 256 threads cover the 192x192 (t,q) plane in float4s.
// qk/kk rows are re-read 192x per head -> keep them regular-temporal so they
// stay L2-resident; the write-once output stream goes out with TH=NT.
// ---------------------------------------------------------------------------
__global__ __launch_bounds__(256) void k_score(
    const float* __restrict__ qk, const float* __restrict__ kk,
    float* __restrict__ out) {
  const int ns = blockIdx.x;
  const int head = ns / SQ, s = ns - head * SQ;
  const float* kkh = kk + (size_t)head * SQ * SQ;
  const float* qkh = qk + (size_t)head * SQ * SQ;
  float* base = out + (size_t)ns * (SQ * SQ);
  for (int idx = threadIdx.x; idx < (SQ * SQ) / 4; idx += 256) {
    const int t = idx / (SQ / 4);
    const int q4 = (idx - t * (SQ / 4)) * 4;
    v4f val;
    if (s > t) {
      val = (v4f){MASK_FILL, MASK_FILL, MASK_FILL, MASK_FILL};
    } else {
      const float kkv = kkh[s * SQ + t];
      const v4f qr = *(const v4f*)(qkh + t * SQ + q4);  // global_load_b128
#pragma unroll
      for (int j = 0; j < 4; ++j) {
        const int qi = q4 + j;
        val[j] = (t > qi) ? MASK_FILL : (kkv + qr[j]);
      }
    }
    __builtin_nontemporal_store(val, (v4f*)(base + t * SQ + q4));  // b128 NT
  }
}

// ---------------------------------------------------------------------------
// Kernel 7: stream v_gated (75 MB): out[n,s,t,h] = silu(v1)[s,h] * v2[t,h].
// One block per (n,s); silu row staged in LDS, non-temporal b128 stores.
// ---------------------------------------------------------------------------
__global__ __launch_bounds__(256) void k_vgated(
    const float* __restrict__ SV, const float* __restrict__ v2,
    float* __restrict__ out) {
  const int ns = blockIdx.x;
  const int head = ns / SQ, s = ns - head * SQ;
  const float* SVh = SV + (size_t)head * SQ * HD;
  const float* v2h = v2 + (size_t)head * SQ * HD;
  float* base = out + (size_t)ns * (SQ * HD);
  __shared__ __align__(16) float sv[HD];
  if (threadIdx.x < HD) sv[threadIdx.x] = SVh[s * HD + threadIdx.x];
  __syncthreads();
  for (int idx = threadIdx.x; idx < (SQ * HD) / 4; idx += 256) {
    const int t = idx / (HD / 4);
    const int h4 = (idx - t * (HD / 4)) * 4;
    v4f a = *(const v4f*)(v2h + t * HD + h4);
    v4f g = *(const v4f*)(sv + h4);
    v4f r = a * g;
    __builtin_nontemporal_store(r, (v4f*)(base + t * HD + h4));  // b128 NT
  }
}

// ---------------------------------------------------------------------------
extern "C" void kernel_launch(void* const* d_in, const int* in_sizes, int n_in,
                              void* d_out, int out_size, void* d_ws,
                              size_t ws_size, hipStream_t stream) {
  const float* q  = (const float*)d_in[0];
  const float* k1 = (const float*)d_in[1];
  const float* k2 = (const float*)d_in[2];
  const float* v1 = (const float*)d_in[3];
  const float* v2 = (const float*)d_in[4];

  float* out = (float*)d_out;
  // Outputs concatenated flat in return order:
  float* z_out  = out;                              //  98304 floats
  float* sc_out = out + 98304;                      //  56623104 floats
  float* vg_out = out + 98304 + 56623104;           //  18874368 floats
  float* M_out  = out + 98304 + 56623104 + 18874368;//  1536 floats

  // Workspace layout (floats). Total ~1.38M floats (~5.6 MB).
  float* ws  = (float*)d_ws;
  float* qk  = ws;                 // 8*192*192
  float* kk  = qk + NH * SQ * SQ;  // 8*192*192
  float* E   = kk + NH * SQ * SQ;  // 8*192*192
  float* P   = E  + NH * SQ * SQ;  // 8*192*192
  float* SV  = P  + NH * SQ * SQ;  // 8*192*64  (silu(v1))
  float* G   = SV + NH * SQ * HD;  // 8*192*64
  float* mkk = G  + NH * SQ * HD;  // 8*192
  float* M0  = mkk + NH * SQ;      // 8*192
  float* SE  = M0 + NH * SQ;       // 8*192

  // 1) qk / kk GEMMs (WMMA f32 16x16x4), one wave per 16x16 tile.
  k_gemm_scores<<<dim3(SQ / 16, SQ / 16, NH * 2), 32, 0, stream>>>(q, k1, k2,
                                                                   qk, kk);
  // 2) softmax statistics + silu(v1) + M output.
  k_stats<<<NH, SQ, 0, stream>>>(v1, qk, kk, SV, mkk, M0, SE, M_out);
  // 3) exp-factor matrices E, P.
  k_pe<<<NH, 256, 0, stream>>>(qk, kk, mkk, M0, E, P);
  // 4) G = v2 * (E^T . silu_v1)   (WMMA, K=192).
  k_gemm_w<<<dim3(SQ / 16, HD / 16, NH), 32, 0, stream>>>(E, SV, v2, G);
  // 5) z = (P . G) / SE           (WMMA, K=192).
  k_gemm_z<<<dim3(SQ / 16, HD / 16, NH), 32, 0, stream>>>(P, G, SE, z_out);
  // 6) stream the 226 MB score tensor (non-temporal stores).
  k_score<<<NH * SQ, 256, 0, stream>>>(qk, kk, sc_out);
  // 7) stream the 75 MB v_gated tensor (non-temporal stores).
  k_vgated<<<NH * SQ, 256, 0, stream>>>(SV, v2, vg_out);
}